// DataEmbedding_cycle_pos_90271622627786
// MI455X (gfx1250) — compile-verified
//
#include <hip/hip_runtime.h>
#include <hip/hip_bf16.h>
#include <math.h>

// ---------------------------------------------------------------------------
// Problem constants (from reference): B=16, T=2048, N=8, D=512
// out[b,t,d] = conv(x)[b,t,d] + temp[b,t,d] + cyc[b,t,d]
// ---------------------------------------------------------------------------
#define TB 16
#define TT 2048
#define TN 8
#define TD 512

typedef __attribute__((ext_vector_type(2))) float v2f;
typedef __attribute__((ext_vector_type(8))) float v8f;

// -log2(10000)/512 : div[j] = exp2(C2LOG * (2j)),  (d & ~1) == 2j
#define C2LOG (-0.025952563262f)

// ===========================================================================
// Kernel 1: per-(b,n) DFT magnitude argmax -> period  (128 series)
// grid = 128 blocks, block = 1024 threads; bin k handled by thread k (+1024)
// Rotation recurrence for cos/sin with exact __sincosf refresh every 64 steps
// (angle reduced mod 2*pi via (k*t)&2047 so large-arg precision is fine).
// ===========================================================================
__global__ void __launch_bounds__(1024)
cycle_period_kernel(const float* __restrict__ x, float* __restrict__ periods)
{
    __shared__ float xv[TT];
    __shared__ float sval[1024];
    __shared__ int   sidx[1024];

    const int bn  = blockIdx.x;          // 0..127
    const int b   = bn >> 3;
    const int n   = bn & 7;
    const int tid = threadIdx.x;

    for (int t = tid; t < TT; t += 1024)
        xv[t] = x[((size_t)b * TT + t) * TN + n];
    __syncthreads();

    const float theta = 3.14159265358979f / 1024.0f;   // 2*pi / T
    float best = -1.0f;
    int   bidx = 0;

    for (int k = tid; k < (TT / 2 + 1); k += 1024) {
        float s0, c0;
        __sincosf(theta * (float)k, &s0, &c0);
        float re = 0.0f, im = 0.0f;
        float c = 1.0f, s = 0.0f;
        for (int t = 0; t < TT; ++t) {
            if ((t & 63) == 0) {                       // exact refresh
                int m = (k * t) & (TT - 1);            // angle mod 2*pi
                __sincosf(theta * (float)m, &s, &c);
            }
            float xt = xv[t];
            re = fmaf(xt, c, re);
            im = fmaf(xt, s, im);
            float cn = c * c0 - s * s0;                // rotate by theta*k
            s = fmaf(s, c0, c * s0);
            c = cn;
        }
        float mag = fmaf(re, re, im * im);
        if (mag > best) { best = mag; bidx = k; }      // k increasing: first-max kept
    }

    sval[tid] = best;
    sidx[tid] = bidx;
    __syncthreads();
    for (int off = 512; off > 0; off >>= 1) {
        if (tid < off) {
            float v2 = sval[tid + off];
            int   i2 = sidx[tid + off];
            if (v2 > sval[tid] || (v2 == sval[tid] && i2 < sidx[tid])) {
                sval[tid] = v2; sidx[tid] = i2;
            }
        }
        __syncthreads();
    }
    if (tid == 0) {
        int   k = sidx[0];
        // jnp.fft.fftfreq(2048): f[k]=k/2048 for k<1024, f[1024]=-0.5
        float f = (k == TT / 2) ? -0.5f : (float)k / (float)TT;
        float p = (f == 0.0f) ? (float)TT : (float)TT / f;
        p = fminf((float)TT, fmaxf(1.0f, p));
        periods[bn] = p;
    }
}

// ===========================================================================
// Kernel 2: fused conv(WMMA f32 16x16x4) + temporal + cycle embedding
// grid = B*T/16 = 2048 blocks, block = 256 threads (8 waves of 32).
// Workgroup tile: 16 time rows x 512 D cols; wave w owns cols [w*64, w*64+64).
// Conv GEMM: A[16 x 24] = x rows (with circular halo), B[24 x 16] = conv_w
// column tile, K split into 6 steps of 4 on V_WMMA_F32_16X16X4_F32.
// ===========================================================================
__global__ void __launch_bounds__(256)
data_embedding_fused_kernel(const float* __restrict__ x,
                            const int*   __restrict__ x_mark,
                            const float* __restrict__ conv_w,
                            const float* __restrict__ periods,
                            float*       __restrict__ out)
{
    __shared__ float wlds[TD * 24];      // 48 KB: conv_w [512][8][3] flat
    __shared__ float xs[18][TN];         // 16 rows + circular halo
    __shared__ float pidx[16][TN];       // floor(t mod period[b,n])
    __shared__ int   xm[16][4];          // x_mark rows

    const int blk = blockIdx.x;
    const int b   = blk >> 7;            // / (T/16)
    const int t0  = (blk & 127) << 4;
    const int tid = threadIdx.x;

    for (int i = tid; i < TD * 24; i += 256)
        wlds[i] = conv_w[i];

    for (int i = tid; i < 18 * TN; i += 256) {
        int r = i >> 3, n = i & 7;
        int t = (t0 + r - 1 + TT) & (TT - 1);            // circular pad
        xs[r][n] = x[((size_t)b * TT + t) * TN + n];
    }
    if (tid < 64) {
        int m = tid >> 2, c = tid & 3;
        xm[m][c] = x_mark[((size_t)b * TT + t0 + m) * 4 + c];
    }
    if (tid < 128) {
        int m = tid >> 3, n = tid & 7;
        float p = periods[b * TN + n];
        pidx[m][n] = floorf(fmodf((float)(t0 + m), p));
    }
    __syncthreads();

    const int wave = tid >> 5;
    const int lane = tid & 31;
    const int half = lane >> 4;          // 0: lanes 0-15, 1: lanes 16-31
    const int c    = lane & 15;

    // --- A fragments (shared by the wave's 4 column tiles) ---------------
    // A[m][kappa] = xs[m + kappa%3][kappa/3]   (kappa = n*3 + dk, K = 24)
    // f32 16x4 layout: lane m = lane&15; VGPR v holds K = k0 + 2*half + v
    v2f afrag[6];
#pragma unroll
    for (int ks = 0; ks < 6; ++ks) {
        int k0 = ks * 4 + 2 * half;
        int k1 = k0 + 1;
        afrag[ks].x = xs[c + (k0 % 3)][k0 / 3];
        afrag[ks].y = xs[c + (k1 % 3)][k1 / 3];
    }

    // --- GEMM: 4 column tiles x 6 K-steps of V_WMMA_F32_16X16X4_F32 ------
    v8f acc[4];
#pragma unroll
    for (int dt = 0; dt < 4; ++dt) {
        int d0 = wave * 64 + dt * 16;
        v8f a = {};
#pragma unroll
        for (int ks = 0; ks < 6; ++ks) {
            int kk = ks * 4 + 2 * half;
            v2f bfrag;                               // B[kappa][col]: col = c
            bfrag.x = wlds[(d0 + c) * 24 + kk];
            bfrag.y = wlds[(d0 + c) * 24 + kk + 1];
            a = __builtin_amdgcn_wmma_f32_16x16x4_f32(
                    /*neg_a=*/false, afrag[ks],
                    /*neg_b=*/false, bfrag,
                    /*c_mod=*/(short)0, a,
                    /*reuse_a=*/false, /*reuse_b=*/false);
        }
        acc[dt] = a;
    }

    // --- Epilogue: temporal + cycle embeddings, write out ----------------
    // C/D layout: VGPR r, lane l -> (M = r + 8*(l>=16), N = l&15)
#pragma unroll
    for (int dt = 0; dt < 4; ++dt) {
        int   d    = wave * 64 + dt * 16 + c;
        float div  = exp2f(C2LOG * (float)(d & ~1));
        bool  evn  = (d & 1) == 0;
#pragma unroll
        for (int r = 0; r < 8; ++r) {
            int m = r + 8 * half;
            int t = t0 + m;
            float emb;
            if (evn) {
                emb = __sinf((float)xm[m][0] * div) + __sinf((float)xm[m][1] * div)
                    + __sinf((float)xm[m][2] * div) + __sinf((float)xm[m][3] * div);
                float s = 0.0f;
#pragma unroll
                for (int n = 0; n < TN; ++n) s += __sinf(pidx[m][n] * div);
                emb = fmaf(s, 0.125f, emb);
            } else {
                emb = __cosf((float)xm[m][0] * div) + __cosf((float)xm[m][1] * div)
                    + __cosf((float)xm[m][2] * div) + __cosf((float)xm[m][3] * div);
                float s = 0.0f;
#pragma unroll
                for (int n = 0; n < TN; ++n) s += __cosf(pidx[m][n] * div);
                emb = fmaf(s, 0.125f, emb);
            }
            out[((size_t)b * TT + t) * TD + d] = acc[dt][r] + emb;
        }
    }
}

// ===========================================================================
// Launch: kernel 1 (periods -> d_ws) then kernel 2 (fused), stream-ordered.
// ===========================================================================
extern "C" void kernel_launch(void* const* d_in, const int* in_sizes, int n_in,
                              void* d_out, int out_size, void* d_ws, size_t ws_size,
                              hipStream_t stream)
{
    const float* x      = (const float*)d_in[0];   // [16,2048,8]  f32
    const int*   x_mark = (const int*)  d_in[1];   // [16,2048,4]  i32
    const float* conv_w = (const float*)d_in[2];   // [512,8,3]    f32
    float*       out    = (float*)d_out;           // [16,2048,512] f32
    float*       per    = (float*)d_ws;            // 128 floats scratch

    hipLaunchKernelGGL(cycle_period_kernel, dim3(TB * TN), dim3(1024), 0, stream,
                       x, per);
    hipLaunchKernelGGL(data_embedding_fused_kernel, dim3(TB * TT / 16), dim3(256),
                       0, stream, x, x_mark, conv_w, per, out);
}